// DualScaleAttentionFusionModule_58695023067823
// MI455X (gfx1250) — compile-verified
//
#include <hip/hip_runtime.h>
#include <math.h>
#include <stdint.h>

typedef __attribute__((ext_vector_type(2))) float v2f;
typedef __attribute__((ext_vector_type(8))) float v8f;

constexpr int B  = 4;
constexpr int Cx = 64;
constexpr int Cy = 128;
constexpr int Hx = 64, Wx = 64;
constexpr int Hy = 32, Wy = 32;
constexpr int Nx = Hx * Wx;   // 4096
constexpr int Ny = Hy * Wy;   // 1024
constexpr int Dd = 8;         // attention head dim
constexpr int R  = 32;        // SE reduction channels

// ---------------------------------------------------------------------------
// 1. Bilinear upsample Y (B,Cy,32,32) -> Y_up (B,Cy,64,64), half-pixel centers
// ---------------------------------------------------------------------------
__global__ void k_upsample(const float* __restrict__ Y, float* __restrict__ yup) {
  int idx = blockIdx.x * 256 + threadIdx.x;       // B*Cy*Nx
  if (idx >= B * Cy * Nx) return;
  int w  = idx & 63;
  int h  = (idx >> 6) & 63;
  int bc = idx >> 12;                             // b*Cy + c
  float sy = fminf(fmaxf(h * 0.5f - 0.25f, 0.f), (float)(Hy - 1));
  float sx = fminf(fmaxf(w * 0.5f - 0.25f, 0.f), (float)(Wy - 1));
  int   y0 = (int)sy, x0 = (int)sx;
  float wy = sy - (float)y0, wxf = sx - (float)x0;
  int   y1 = min(y0 + 1, Hy - 1), x1 = min(x0 + 1, Wy - 1);
  const float* src = Y + bc * Ny;
  float v00 = src[y0 * Wy + x0], v01 = src[y0 * Wy + x1];
  float v10 = src[y1 * Wy + x0], v11 = src[y1 * Wy + x1];
  yup[idx] = (1.f - wy) * ((1.f - wxf) * v00 + wxf * v01)
           +        wy  * ((1.f - wxf) * v10 + wxf * v11);
}

// ---------------------------------------------------------------------------
// 2. 3x3 conv on Y_up producing 18 offset channels, zero padding
// ---------------------------------------------------------------------------
__global__ void k_offset_conv(const float* __restrict__ yup, const float* __restrict__ ow,
                              const float* __restrict__ ob, float* __restrict__ offs) {
  int idx = blockIdx.x * 256 + threadIdx.x;       // B*18*Nx = 294912
  if (idx >= B * 18 * Nx) return;
  int w = idx & 63, h = (idx >> 6) & 63;
  int o = (idx >> 12) % 18;
  int b = idx / (18 * Nx);
  const float* yb = yup + b * Cy * Nx;
  const float* wt = ow + o * Cy * 9;
  float acc = ob[o];
  for (int c = 0; c < Cy; ++c) {
    const float* yc = yb + c * Nx;
    const float* wc = wt + c * 9;
#pragma unroll
    for (int i = 0; i < 3; ++i) {
      int hy = h + i - 1;
      if ((unsigned)hy >= 64u) continue;
#pragma unroll
      for (int j = 0; j < 3; ++j) {
        int wx2 = w + j - 1;
        if ((unsigned)wx2 >= 64u) continue;
        acc += yc[hy * 64 + wx2] * wc[i * 3 + j];
      }
    }
  }
  offs[idx] = acc;
}

// ---------------------------------------------------------------------------
// 3. Deformable 3x3 conv (Cy=128 -> Cx=64). Block = 256 threads = 4 pixels x
//    64 out-channels. Phase 1: bilinear-sample im2col tile (4 x 1152 f32,
//    18.4 KB LDS). Phase 2: each thread reduces 1152 MACs for one (pix,oc).
// ---------------------------------------------------------------------------
__global__ void k_deform(const float* __restrict__ yup, const float* __restrict__ offs,
                         const float* __restrict__ dw, const float* __restrict__ db,
                         float* __restrict__ yal) {
  __shared__ float smp[4][Cy * 9];
  int p = threadIdx.x >> 6;                       // pixel slot 0..3
  int t = threadIdx.x & 63;                       // 0..63
  int pix = blockIdx.x * 4 + p;                   // over B*Nx
  int b = pix >> 12;
  int n = pix & (Nx - 1);
  int h = n >> 6, w = n & 63;
  const float* yb  = yup  + b * Cy * Nx;
  const float* ofb = offs + b * 18 * Nx;
  for (int e = t; e < Cy * 9; e += 64) {
    int k = e >> 7;                               // 0..8
    int c = e & 127;
    float dy = ofb[(2 * k + 0) * Nx + n];
    float dx = ofb[(2 * k + 1) * Nx + n];
    float py = (float)h - 1.f + (float)(k / 3) + dy;
    float px = (float)w - 1.f + (float)(k % 3) + dx;
    float y0f = floorf(py), x0f = floorf(px);
    float wy = py - y0f, wxf = px - x0f;
    int y0 = (int)y0f, x0 = (int)x0f;
    const float* yc = yb + c * Nx;
    float v00 = ((unsigned)y0       < 64u && (unsigned)x0       < 64u) ? yc[y0 * 64 + x0]           : 0.f;
    float v01 = ((unsigned)y0       < 64u && (unsigned)(x0 + 1) < 64u) ? yc[y0 * 64 + x0 + 1]       : 0.f;
    float v10 = ((unsigned)(y0 + 1) < 64u && (unsigned)x0       < 64u) ? yc[(y0 + 1) * 64 + x0]     : 0.f;
    float v11 = ((unsigned)(y0 + 1) < 64u && (unsigned)(x0 + 1) < 64u) ? yc[(y0 + 1) * 64 + x0 + 1] : 0.f;
    smp[p][e] = (1.f - wy) * ((1.f - wxf) * v00 + wxf * v01)
              +        wy  * ((1.f - wxf) * v10 + wxf * v11);
  }
  __syncthreads();
  float acc = db[t];
  const float* wt = dw + t * Cy * 9;              // (o, c, k)
  for (int c = 0; c < Cy; ++c) {
#pragma unroll
    for (int k = 0; k < 9; ++k)
      acc += smp[p][k * 128 + c] * wt[c * 9 + k];
  }
  yal[(b * Cx + t) * Nx + n] = acc;
}

// ---------------------------------------------------------------------------
// 4. Generic 1x1 conv (channel GEMV per pixel) with bias
// ---------------------------------------------------------------------------
__global__ void k_conv1x1(const float* __restrict__ in, const float* __restrict__ w,
                          const float* __restrict__ bias, float* __restrict__ out,
                          int Cin, int Cout, int N) {
  int idx = blockIdx.x * 256 + threadIdx.x;       // B*Cout*N
  if (idx >= B * Cout * N) return;
  int n = idx % N;
  int o = (idx / N) % Cout;
  int b = idx / (N * Cout);
  const float* ib = in + b * Cin * N + n;
  const float* wr = w + o * Cin;
  float acc = bias[o];
  for (int c = 0; c < Cin; ++c) acc += ib[c * N] * wr[c];
  out[idx] = acc;
}

// ---------------------------------------------------------------------------
// 5. QK^T logits, one wave per 16x16 tile.  D[n,m] = sum_c Q[c,n]*K[c,m]
//    f32 WMMA 16x16x4, K-dim = 8 -> 2 steps.
// ---------------------------------------------------------------------------
__global__ void k_qk_wmma(const float* __restrict__ Q, const float* __restrict__ Km,
                          float* __restrict__ attn) {
  int lane = threadIdx.x;                         // block = 32 (one wave)
  int m0 = blockIdx.x * 16;                       // key tile
  int n0 = blockIdx.y * 16;                       // query tile
  int half = lane >> 4;
  int l = lane & 15;
  v8f acc = {};
#pragma unroll
  for (int k0 = 0; k0 < Dd; k0 += 4) {
    int c = k0 + 2 * half;
    v2f a, bf;
    a.x  = Q[(c + 0) * Nx + n0 + l];
    a.y  = Q[(c + 1) * Nx + n0 + l];
    bf.x = Km[(c + 0) * Ny + m0 + l];
    bf.y = Km[(c + 1) * Ny + m0 + l];
    acc = __builtin_amdgcn_wmma_f32_16x16x4_f32(false, a, false, bf,
                                                (short)0, acc, false, false);
  }
#pragma unroll
  for (int r = 0; r < 8; ++r) {
    int n = n0 + r + 8 * half;                    // C/D layout: M = r + 8*half
    attn[n * Ny + m0 + l] = acc[r];
  }
}

// ---------------------------------------------------------------------------
// 6. Row softmax over 1024 keys, one wave32 per row (8 rows / 256-thr block)
// ---------------------------------------------------------------------------
__global__ void k_softmax(float* __restrict__ attn) {
  int row  = blockIdx.x * 8 + (threadIdx.x >> 5);
  int lane = threadIdx.x & 31;
  float* rp = attn + row * Ny;
  float mx = -1e30f;
  for (int j = lane; j < Ny; j += 32) mx = fmaxf(mx, rp[j]);
  for (int s = 16; s > 0; s >>= 1) mx = fmaxf(mx, __shfl_xor(mx, s, 32));
  float sum = 0.f;
  for (int j = lane; j < Ny; j += 32) { float e = __expf(rp[j] - mx); rp[j] = e; sum += e; }
  for (int s = 16; s > 0; s >>= 1) sum += __shfl_xor(sum, s, 32);
  float inv = 1.f / sum;
  for (int j = lane; j < Ny; j += 32) rp[j] *= inv;
}

// ---------------------------------------------------------------------------
// 7. AV GEMM: D[c,n] = sum_m V[c,m]*attn[n,m]  (M=64, N=4096, K=1024).
//    Block = 128 threads (4 waves) per 16-column n-tile; the shared 16x1024
//    attn B-tile is staged into LDS ONCE via async global->LDS (ASYNCcnt
//    path), then all 4 waves (one 16-row c-tile each) stream B-fragments
//    from LDS. Fused epilogue: Z = gamma*D + X + Y_al.
//    LDS rows padded to 1028 floats: rows stay 16B aligned for b128 async
//    writes and the 16 reader lanes land on distinct banks (stride 4 banks).
// ---------------------------------------------------------------------------
__global__ void k_av_wmma(const float* __restrict__ V, const float* __restrict__ attn,
                          const float* __restrict__ Xb, const float* __restrict__ Yalb,
                          const float* __restrict__ gamma, float* __restrict__ Zb) {
  __shared__ float bt[16][1028];
  int tid = threadIdx.x;                          // 128 threads = 4 waves
  int n0 = blockIdx.x * 16;

  // ---- async stage: attn[n0..n0+15][0..1023] -> bt ----
  {
    const float* gbase = attn + (size_t)n0 * Ny;  // block-uniform -> SGPR pair
    uint32_t lbase = (uint32_t)(uintptr_t)(&bt[0][0]);
    int r  = tid >> 3;                            // row 0..15
    int k0 = tid & 7;                             // starting 16B chunk
    for (int k = k0; k < 256; k += 8) {           // 32 x b128 per thread
      uint32_t loff = lbase + (uint32_t)((r * 1028 + 4 * k) * 4);
      uint32_t goff = (uint32_t)((r * Ny + 4 * k) * 4);
      asm volatile("global_load_async_to_lds_b128 %0, %1, %2"
                   :: "v"(loff), "v"(goff), "s"(gbase) : "memory");
    }
    asm volatile("s_wait_asynccnt 0x0" ::: "memory");
  }
  __syncthreads();

  int lane = tid & 31;
  int c0   = (tid >> 5) * 16;                     // wave -> c-tile
  int half = lane >> 4;
  int l    = lane & 15;
  v8f acc = {};
  for (int k0 = 0; k0 < Ny; k0 += 4) {
    int kk = k0 + 2 * half;
    v2f a, bf;
    a.x  = V[(c0 + l) * Ny + kk + 0];             // A[M=c][K=m] : b64 load
    a.y  = V[(c0 + l) * Ny + kk + 1];
    bf.x = bt[l][kk + 0];                         // B[K=m][N=n] from LDS
    bf.y = bt[l][kk + 1];
    acc = __builtin_amdgcn_wmma_f32_16x16x4_f32(false, a, false, bf,
                                                (short)0, acc, false, false);
  }
  float g = gamma[0];
#pragma unroll
  for (int r = 0; r < 8; ++r) {
    int c = c0 + r + 8 * half;
    int o = c * Nx + n0 + l;
    Zb[o] = g * acc[r] + Xb[o] + Yalb[o];
  }
}

// ---------------------------------------------------------------------------
// 8. Per-(b,c) spatial mean of Z  -> s (B*Cx)
// ---------------------------------------------------------------------------
__global__ void k_rowmean(const float* __restrict__ Z, float* __restrict__ s) {
  __shared__ float red[256];
  int bc = blockIdx.x;
  const float* zp = Z + bc * Nx;
  float acc = 0.f;
  for (int j = threadIdx.x; j < Nx; j += 256) acc += zp[j];
  red[threadIdx.x] = acc; __syncthreads();
  for (int st = 128; st > 0; st >>= 1) {
    if (threadIdx.x < st) red[threadIdx.x] += red[threadIdx.x + st];
    __syncthreads();
  }
  if (threadIdx.x == 0) s[bc] = red[0] * (1.f / (float)Nx);
}

// ---------------------------------------------------------------------------
// 9. SE MLP: x1 = sigmoid(fc2 @ relu(fc1 @ s)), single block
// ---------------------------------------------------------------------------
__global__ void k_se_mlp(const float* __restrict__ s, const float* __restrict__ fc1,
                         const float* __restrict__ fc2, float* __restrict__ x1) {
  __shared__ float hsh[B][R];
  int tid = threadIdx.x;                          // 256
  if (tid < B * R) {
    int b = tid / R, j = tid % R;
    float acc = 0.f;
    for (int c = 0; c < Cx; ++c) acc += fc1[j * Cx + c] * s[b * Cx + c];
    hsh[b][j] = fmaxf(acc, 0.f);
  }
  __syncthreads();
  int b = tid / Cx, o = tid % Cx;                 // B*Cx == 256
  float acc = 0.f;
  for (int j = 0; j < R; ++j) acc += fc2[o * R + j] * hsh[b][j];
  x1[tid] = 1.f / (1.f + __expf(-acc));
}

// ---------------------------------------------------------------------------
// 10. BatchNorm batch statistics: sum & sumsq per channel (one block/channel)
// ---------------------------------------------------------------------------
__global__ void k_bnstats(const float* __restrict__ x, float* __restrict__ stats,
                          int C, int N) {
  __shared__ float r1[256], r2[256];
  int ch = blockIdx.x;
  float s1 = 0.f, s2 = 0.f;
  for (int i = threadIdx.x; i < B * N; i += 256) {
    int b = i / N, n = i % N;
    float v = x[(b * C + ch) * N + n];
    s1 += v; s2 += v * v;
  }
  r1[threadIdx.x] = s1; r2[threadIdx.x] = s2; __syncthreads();
  for (int st = 128; st > 0; st >>= 1) {
    if (threadIdx.x < st) {
      r1[threadIdx.x] += r1[threadIdx.x + st];
      r2[threadIdx.x] += r2[threadIdx.x + st];
    }
    __syncthreads();
  }
  if (threadIdx.x == 0) { stats[ch] = r1[0]; stats[C + ch] = r2[0]; }
}

// ---------------------------------------------------------------------------
// 11. In-place BN + ReLU
// ---------------------------------------------------------------------------
__global__ void k_bn_relu(float* __restrict__ x, const float* __restrict__ stats,
                          const float* __restrict__ wgt, const float* __restrict__ bia,
                          int C, int N) {
  int idx = blockIdx.x * 256 + threadIdx.x;
  if (idx >= B * C * N) return;
  int ch = (idx / N) % C;
  float cnt = (float)(B * N);
  float m   = stats[ch] / cnt;
  float var = stats[C + ch] / cnt - m * m;
  float inv = rsqrtf(var + 1e-5f);
  float v = (x[idx] - m) * inv * wgt[ch] + bia[ch];
  x[idx] = fmaxf(v, 0.f);
}

// ---------------------------------------------------------------------------
// 12. Final fused gate: x2=bn2(x2b); x3=sigmoid(g1*x1+g2*x2); Zg=Z*x3;
//     out = X*Zg + Y_al*(1-Zg)
// ---------------------------------------------------------------------------
__global__ void k_final(const float* __restrict__ X, const float* __restrict__ Z,
                        const float* __restrict__ Yal, const float* __restrict__ x2b,
                        const float* __restrict__ bn2s, const float* __restrict__ bn2w,
                        const float* __restrict__ bn2b, const float* __restrict__ x1,
                        const float* __restrict__ g1, const float* __restrict__ g2,
                        float* __restrict__ out) {
  int idx = blockIdx.x * 256 + threadIdx.x;       // B*Cx*Nx
  if (idx >= B * Cx * Nx) return;
  int bc = idx / Nx;
  int ch = bc % Cx;
  float cnt = (float)(B * Nx);
  float m   = bn2s[ch] / cnt;
  float var = bn2s[Cx + ch] / cnt - m * m;
  float inv = rsqrtf(var + 1e-5f);
  float x2 = (x2b[idx] - m) * inv * bn2w[ch] + bn2b[ch];
  float x3 = 1.f / (1.f + __expf(-(g1[0] * x1[bc] + g2[0] * x2)));
  float zg = Z[idx] * x3;
  out[idx] = X[idx] * zg + Yal[idx] * (1.f - zg);
}

// ---------------------------------------------------------------------------
extern "C" void kernel_launch(void* const* d_in, const int* in_sizes, int n_in,
                              void* d_out, int out_size, void* d_ws, size_t ws_size,
                              hipStream_t stream) {
  const float* X        = (const float*)d_in[0];
  const float* Y        = (const float*)d_in[1];
  const float* offset_w = (const float*)d_in[2];
  const float* offset_b = (const float*)d_in[3];
  const float* deform_w = (const float*)d_in[4];
  const float* deform_b = (const float*)d_in[5];
  const float* q_w      = (const float*)d_in[6];
  const float* q_b      = (const float*)d_in[7];
  const float* k_w      = (const float*)d_in[8];
  const float* k_b      = (const float*)d_in[9];
  const float* v_w      = (const float*)d_in[10];
  const float* v_b      = (const float*)d_in[11];
  const float* gamma    = (const float*)d_in[12];
  const float* fc1_w    = (const float*)d_in[13];
  const float* fc2_w    = (const float*)d_in[14];
  const float* conv1_w  = (const float*)d_in[15];
  const float* conv1_b  = (const float*)d_in[16];
  const float* bn1_w    = (const float*)d_in[17];
  const float* bn1_b    = (const float*)d_in[18];
  const float* conv2_w  = (const float*)d_in[19];
  const float* conv2_b  = (const float*)d_in[20];
  const float* bn2_w    = (const float*)d_in[21];
  const float* bn2_b    = (const float*)d_in[22];
  const float* gate1    = (const float*)d_in[23];
  const float* gate2    = (const float*)d_in[24];
  float* out = (float*)d_out;

  float* ws   = (float*)d_ws;
  float* YUP  = ws;                    // B*Cy*Nx       = 2097152
  float* OFFS = YUP  + B * Cy * Nx;    // B*18*Nx       =  294912
  float* YAL  = OFFS + B * 18 * Nx;    // B*Cx*Nx       = 1048576
  float* QB   = YAL  + B * Cx * Nx;    // B*Dd*Nx       =  131072
  float* KB   = QB   + B * Dd * Nx;    // B*Dd*Ny       =   32768
  float* VB   = KB   + B * Dd * Ny;    // B*Cx*Ny       =  262144
  float* ATTN = VB   + B * Cx * Ny;    // Nx*Ny (reused)= 4194304
  float* ZB   = ATTN + Nx * Ny;        // B*Cx*Nx       = 1048576
  float* X2A  = ZB   + B * Cx * Nx;    // B*R*Nx        =  524288
  float* X2B  = X2A  + B * R * Nx;     // B*Cx*Nx       = 1048576
  float* SSE  = X2B  + B * Cx * Nx;    // B*Cx          =     256
  float* X1   = SSE  + B * Cx;         // B*Cx          =     256
  float* BN1S = X1   + B * Cx;         // 2*R           =      64
  float* BN2S = BN1S + 2 * R;          // 2*Cx          =     128

  // Upsample + offset conv + deformable conv
  k_upsample<<<(B * Cy * Nx) / 256, 256, 0, stream>>>(Y, YUP);
  k_offset_conv<<<(B * 18 * Nx + 255) / 256, 256, 0, stream>>>(YUP, offset_w, offset_b, OFFS);
  k_deform<<<(B * Nx) / 4, 256, 0, stream>>>(YUP, OFFS, deform_w, deform_b, YAL);

  // Q, K, V projections
  k_conv1x1<<<(B * Dd * Nx) / 256, 256, 0, stream>>>(X, q_w, q_b, QB, Cx, Dd, Nx);
  k_conv1x1<<<(B * Dd * Ny) / 256, 256, 0, stream>>>(Y, k_w, k_b, KB, Cy, Dd, Ny);
  k_conv1x1<<<(B * Cx * Ny) / 256, 256, 0, stream>>>(Y, v_w, v_b, VB, Cy, Cx, Ny);

  // Attention per batch (attn buffer reused -> stays L2 resident)
  for (int b = 0; b < B; ++b) {
    k_qk_wmma<<<dim3(Ny / 16, Nx / 16), 32, 0, stream>>>(QB + b * Dd * Nx,
                                                         KB + b * Dd * Ny, ATTN);
    k_softmax<<<Nx / 8, 256, 0, stream>>>(ATTN);
    k_av_wmma<<<Nx / 16, 128, 0, stream>>>(VB + b * Cx * Ny, ATTN,
                                           X + b * Cx * Nx,
                                           YAL + b * Cx * Nx,
                                           gamma, ZB + b * Cx * Nx);
  }

  // SE branch
  k_rowmean<<<B * Cx, 256, 0, stream>>>(ZB, SSE);
  k_se_mlp<<<1, 256, 0, stream>>>(SSE, fc1_w, fc2_w, X1);

  // conv1 -> BN1 -> ReLU -> conv2 -> BN2 stats
  k_conv1x1<<<(B * R * Nx) / 256, 256, 0, stream>>>(ZB, conv1_w, conv1_b, X2A, Cx, R, Nx);
  k_bnstats<<<R, 256, 0, stream>>>(X2A, BN1S, R, Nx);
  k_bn_relu<<<(B * R * Nx) / 256, 256, 0, stream>>>(X2A, BN1S, bn1_w, bn1_b, R, Nx);
  k_conv1x1<<<(B * Cx * Nx) / 256, 256, 0, stream>>>(X2A, conv2_w, conv2_b, X2B, R, Cx, Nx);
  k_bnstats<<<Cx, 256, 0, stream>>>(X2B, BN2S, Cx, Nx);

  // Final fused gating
  k_final<<<(B * Cx * Nx) / 256, 256, 0, stream>>>(X, ZB, YAL, X2B, BN2S, bn2_w, bn2_b,
                                                   X1, gate1, gate2, out);
}